// ParallelGNNPolicyAC_85676007621124
// MI455X (gfx1250) — compile-verified
//
#include <hip/hip_runtime.h>

// ---------------- problem constants (from reference) ----------------
#define N_NODES 50000
#define N_EDGES 800000
#define IN_DIM  16
#define HID     128

typedef float v2f __attribute__((ext_vector_type(2)));
typedef float v8f __attribute__((ext_vector_type(8)));

// ---------------- degree / normalization prep ----------------

__global__ void k_init_deg(float* __restrict__ deg) {
    int i = blockIdx.x * blockDim.x + threadIdx.x;
    if (i < N_NODES) deg[i] = 1.0f;  // self-loop contributes 1
}

__global__ void k_deg_accum(const long long* __restrict__ dst, float* __restrict__ deg) {
    int e = blockIdx.x * blockDim.x + threadIdx.x;
    if (e < N_EDGES) atomicAdd(&deg[(int)dst[e]], 1.0f);
}

__global__ void k_dis(float* __restrict__ deg) {
    int i = blockIdx.x * blockDim.x + threadIdx.x;
    if (i < N_NODES) {
        float d = deg[i];
        deg[i] = (d > 0.0f) ? rsqrtf(d) : 0.0f;   // in-place: deg -> dis
    }
}

// Repack int64 edge_index -> int32 + precompute norm[e] = dis[s]*dis[d].
// Done once, reused by all 4 convs (cuts edge-stream bandwidth 20B->12B/edge).
__global__ void k_edge_prep(const long long* __restrict__ ei, const float* __restrict__ dis,
                            int* __restrict__ src32, int* __restrict__ dst32,
                            float* __restrict__ normw) {
    int e = blockIdx.x * blockDim.x + threadIdx.x;
    if (e < N_EDGES) {
        int s = (int)ei[e];
        int d = (int)ei[N_EDGES + e];
        src32[e] = s;
        dst32[e] = d;
        normw[e] = dis[s] * dis[d];
    }
}

// ---------------- WMMA GEMM: Out[m][n] = sum_k act(A[m][k]) * W[n][k] ----------------
// fp32 WMMA (V_WMMA_F32_16X16X4_F32): exact-precision match to the fp32 reference;
// the GEMMs are ~3.7 GFLOP total and never the bottleneck here.
// One wave computes a 16(M) x 128(N) tile. N_NODES % 16 == 0 so no M boundary.
template <int K, bool RELU>
__global__ void k_gemm_wmma(const float* __restrict__ A, const float* __restrict__ W,
                            float* __restrict__ Out) {
    const int wid  = (int)((blockIdx.x * blockDim.x + threadIdx.x) >> 5);
    const int lane = (int)(threadIdx.x & 31);
    const int MT   = N_NODES / 16;          // 3125 tiles exactly
    if (wid >= MT) return;                   // wave-uniform: EXEC stays all-ones for WMMA

    const int m0   = wid * 16;
    const int half = lane >> 4;              // 0: lanes 0-15, 1: lanes 16-31
    const int l15  = lane & 15;
    const int row  = m0 + l15;

    v8f acc[8] = {};                         // 8 N-subtiles of 16 cols, fp32 accum

    const float* Arow = A + (long)row * K;
    for (int k0 = 0; k0 < K; k0 += 4) {
        // A 16x4 fp32 fragment: VGPR0=K(k0+2*half), VGPR1=K(k0+2*half+1)  (ISA 7.12.2)
        const int kA = k0 + 2 * half;
        float a0 = Arow[kA];
        float a1 = Arow[kA + 1];
        if (RELU) { a0 = fmaxf(a0, 0.0f); a1 = fmaxf(a1, 0.0f); }
        v2f a; a.x = a0; a.y = a1;

#pragma unroll
        for (int n = 0; n < 8; ++n) {
            // B 4x16 fragment: B[k][col] = W[col][k]  (W is [out,K] row-major)
            const int col = n * 16 + l15;
            const float* Wc = W + (long)col * K + k0 + 2 * half;
            v2f b; b.x = Wc[0]; b.y = Wc[1];
            acc[n] = __builtin_amdgcn_wmma_f32_16x16x4_f32(
                /*neg_a=*/false, a, /*neg_b=*/false, b,
                /*c_mod=*/(short)0, acc[n], /*reuse_a=*/false, /*reuse_b=*/false);
        }
    }

    // C/D layout: VGPR r -> row m0 + r + 8*half, col = n*16 + (lane&15)
#pragma unroll
    for (int n = 0; n < 8; ++n) {
        const int col = n * 16 + l15;
#pragma unroll
        for (int r = 0; r < 8; ++r) {
            const int rr = m0 + r + 8 * half;
            Out[(long)rr * HID + col] = acc[n][r];
        }
    }
}

// ---------------- aggregation ----------------
// Init with self-loop term + bias:  agg[i][f] = dis[i]^2 * lin[i][f] + b[f]
__global__ void k_agg_init(const float* __restrict__ lin, const float* __restrict__ dis,
                           const float* __restrict__ bias, float* __restrict__ agg) {
    int t = blockIdx.x * blockDim.x + threadIdx.x;      // N_NODES * 32 threads (float4 each)
    if (t >= N_NODES * (HID / 4)) return;
    const int nd = t >> 5;
    const int f4 = (t & 31) << 2;
    const float d  = dis[nd];
    const float n2 = d * d;
    const float4 l = *(const float4*)(lin + (long)nd * HID + f4);
    float4 o;
    o.x = n2 * l.x + bias[f4 + 0];
    o.y = n2 * l.y + bias[f4 + 1];
    o.z = n2 * l.z + bias[f4 + 2];
    o.w = n2 * l.w + bias[f4 + 3];
    *(float4*)(agg + (long)nd * HID + f4) = o;
}

// Edge scatter: agg[dst] += norm[e] * lin[src].  float4 gather (L2-resident:
// feature matrix is 25.6 MB << 192 MB L2) + 4 f32 atomics per thread.
__global__ void k_agg_edges(const float* __restrict__ lin, const int* __restrict__ src,
                            const int* __restrict__ dst, const float* __restrict__ normw,
                            float* __restrict__ agg) {
    int t = blockIdx.x * blockDim.x + threadIdx.x;      // N_EDGES * 32 threads
    if (t >= N_EDGES * (HID / 4)) return;
    const int e  = t >> 5;
    const int f4 = (t & 31) << 2;
    const int s  = src[e];
    const int d  = dst[e];
    const float nw = normw[e];
    const float4 l = *(const float4*)(lin + (long)s * HID + f4);
    float* p = agg + (long)d * HID + f4;
    atomicAdd(p + 0, nw * l.x);
    atomicAdd(p + 1, nw * l.y);
    atomicAdd(p + 2, nw * l.z);
    atomicAdd(p + 3, nw * l.w);
}

// ---------------- head: out[n] = dot(relu(agg[n]), Wh) + bh ----------------
__global__ void k_head(const float* __restrict__ agg, const float* __restrict__ Wh,
                       const float* __restrict__ bh, float* __restrict__ out) {
    const int wid  = (int)((blockIdx.x * blockDim.x + threadIdx.x) >> 5);
    const int lane = (int)(threadIdx.x & 31);
    if (wid >= N_NODES) return;
    float p = 0.0f;
#pragma unroll
    for (int j = 0; j < 4; ++j) {
        const int f = lane + j * 32;
        p += fmaxf(agg[(long)wid * HID + f], 0.0f) * Wh[f];
    }
#pragma unroll
    for (int off = 16; off > 0; off >>= 1) p += __shfl_xor(p, off, 32);  // wave32 reduce
    if (lane == 0) out[wid] = p + bh[0];
}

// ---------------- launch ----------------
extern "C" void kernel_launch(void* const* d_in, const int* in_sizes, int n_in,
                              void* d_out, int out_size, void* d_ws, size_t ws_size,
                              hipStream_t stream) {
    (void)in_sizes; (void)n_in; (void)out_size; (void)ws_size;

    const float*     x    = (const float*)d_in[0];
    const long long* ei   = (const long long*)d_in[1];   // int64 per reference
    const float*     W_c1 = (const float*)d_in[2];
    const float*     b_c1 = (const float*)d_in[3];
    const float*     W_c2 = (const float*)d_in[4];
    const float*     b_c2 = (const float*)d_in[5];
    const float*     W_p  = (const float*)d_in[6];
    const float*     b_p  = (const float*)d_in[7];
    const float*     W_v1 = (const float*)d_in[8];
    const float*     b_v1 = (const float*)d_in[9];
    const float*     W_v2 = (const float*)d_in[10];
    const float*     b_v2 = (const float*)d_in[11];
    const float*     W_vh = (const float*)d_in[12];
    const float*     b_vh = (const float*)d_in[13];

    float* out = (float*)d_out;   // [0,N): action_mean, [N,2N): value

    // workspace carve-up (256B aligned)
    char*  ws  = (char*)d_ws;
    size_t off = 0;
    auto carve = [&](size_t bytes) -> void* {
        void* p = ws + off;
        off += bytes;
        off = (off + 255) & ~(size_t)255;
        return p;
    };
    float* dis   = (float*)carve((size_t)N_NODES * 4);   // deg, then rsqrt(deg) in place
    int*   src32 = (int*)  carve((size_t)N_EDGES * 4);
    int*   dst32 = (int*)  carve((size_t)N_EDGES * 4);
    float* normw = (float*)carve((size_t)N_EDGES * 4);
    float* lin   = (float*)carve((size_t)N_NODES * HID * 4);
    float* agg   = (float*)carve((size_t)N_NODES * HID * 4);

    const int TB = 256;
    const int gNode    = (N_NODES + TB - 1) / TB;                // 196
    const int gEdge    = (N_EDGES + TB - 1) / TB;                // 3125
    const int gGemm    = ((N_NODES / 16) + 8 - 1) / 8;           // 3125 waves, 8/block
    const int gAggInit = (N_NODES * (HID / 4) + TB - 1) / TB;    // 6250
    const int gAggEdge = (N_EDGES * (HID / 4) + TB - 1) / TB;    // 100000
    const int gHead    = (N_NODES + 8 - 1) / 8;                  // 6250 (wave per node)

    // normalization prep (shared by both branches)
    k_init_deg <<<gNode, TB, 0, stream>>>(dis);
    k_deg_accum<<<gEdge, TB, 0, stream>>>(ei + N_EDGES, dis);
    k_dis      <<<gNode, TB, 0, stream>>>(dis);
    k_edge_prep<<<gEdge, TB, 0, stream>>>(ei, dis, src32, dst32, normw);

    auto run_branch = [&](const float* W1, const float* b1, const float* W2, const float* b2,
                          const float* Wh, const float* bh, float* outp) {
        // layer 1: lin = x @ W1^T ; agg = norm-aggregate(lin) + b1
        k_gemm_wmma<IN_DIM, false><<<gGemm, TB, 0, stream>>>(x, W1, lin);
        k_agg_init <<<gAggInit, TB, 0, stream>>>(lin, dis, b1, agg);
        k_agg_edges<<<gAggEdge, TB, 0, stream>>>(lin, src32, dst32, normw, agg);
        // layer 2: lin = relu(agg) @ W2^T ; agg = norm-aggregate(lin) + b2
        k_gemm_wmma<HID, true><<<gGemm, TB, 0, stream>>>(agg, W2, lin);
        k_agg_init <<<gAggInit, TB, 0, stream>>>(lin, dis, b2, agg);
        k_agg_edges<<<gAggEdge, TB, 0, stream>>>(lin, src32, dst32, normw, agg);
        // head: outp[n] = dot(relu(agg[n]), Wh) + bh
        k_head     <<<gHead, TB, 0, stream>>>(agg, Wh, bh, outp);
    };

    run_branch(W_c1, b_c1, W_c2, b_c2, W_p,  b_p,  out);            // action_mean
    run_branch(W_v1, b_v1, W_v2, b_v2, W_vh, b_vh, out + N_NODES);  // value
}